// DescriptorContrastiveLoss_45715631899326
// MI455X (gfx1250) — compile-verified
//
#include <hip/hip_runtime.h>
#include <hip/hip_bf16.h>
#include <float.h>

typedef __attribute__((ext_vector_type(2))) float v2f;
typedef __attribute__((ext_vector_type(8))) float v8f;

#define CDNA5_EPS 1e-8f

// ---------------------------------------------------------------------------
// Antialiased trilinear resize (matches jax.image.resize(..., antialias=True)
// for integer downscale): per-axis triangle kernel of radius `scale`,
// 2*scale taps, OOB taps masked and weights renormalized per axis.
// Output is augmented for the WMMA distance trick:
//   augA[n] = ( a.x, a.y, a.z, 1 )
//   augB[n] = ( -2 b.x, -2 b.y, -2 b.z, |b|^2 )
// so that  augA . augB = -2 a.b + |b|^2   (argmin-equivalent to |a-b|^2)
// ---------------------------------------------------------------------------
__device__ inline float axis_weights(int o, int scale, int* i0, float* w) {
    // center of output o in input coords: c = scale*o + scale/2 - 0.5
    // taps i = start + k, k in [0, 2*scale), |i - c| = |k - (scale-0.5)|
    int start = o * scale - scale / 2;
    float cc = (float)scale - 0.5f;
    float sum = 0.f;
    for (int k = 0; k < 2 * scale; ++k) {
        int i = start + k;
        float wk = 1.0f - fabsf((float)k - cc) / (float)scale;
        if (i < 0 || i >= 64) wk = 0.f;  // mask OOB, renormalize below
        w[k] = wk;
        sum += wk;
    }
    *i0 = start;
    return sum;
}

__global__ void resize_aug_kernel(const float* __restrict__ src,
                                  const float* __restrict__ tgt,
                                  float4* __restrict__ augA,
                                  float4* __restrict__ augB,
                                  int S, int scale)
{
    const int IN = 64;
    const size_t IN3 = (size_t)IN * IN * IN;
    int idx = blockIdx.x * blockDim.x + threadIdx.x;
    int N = S * S * S;
    int b = idx / N;
    int n = idx - b * N;
    int x = n % S;
    int y = (n / S) % S;
    int z = n / (S * S);

    float wz[16], wy[16], wx[16];
    int iz0, iy0, ix0;
    float sz = axis_weights(z, scale, &iz0, wz);
    float sy = axis_weights(y, scale, &iy0, wy);
    float sx = axis_weights(x, scale, &ix0, wx);

    float sacc[3] = {0.f, 0.f, 0.f};
    float tacc[3] = {0.f, 0.f, 0.f};
    for (int kz = 0; kz < 2 * scale; ++kz) {
        if (wz[kz] == 0.f) continue;
        for (int ky = 0; ky < 2 * scale; ++ky) {
            if (wy[ky] == 0.f) continue;
            float wzy = wz[kz] * wy[ky];
            size_t rowbase = ((size_t)(iz0 + kz) * IN + (size_t)(iy0 + ky)) * IN;
            for (int kx = 0; kx < 2 * scale; ++kx) {
                float w = wzy * wx[kx];
                if (w == 0.f) continue;
                size_t off = rowbase + (size_t)(ix0 + kx);
#pragma unroll
                for (int c = 0; c < 3; ++c) {
                    size_t p = (size_t)(b * 3 + c) * IN3 + off;
                    sacc[c] += w * src[p];
                    tacc[c] += w * tgt[p];
                }
            }
        }
    }
    float inv = 1.0f / (sz * sy * sx);
    float s0 = sacc[0] * inv, s1 = sacc[1] * inv, s2 = sacc[2] * inv;
    float t0 = tacc[0] * inv, t1 = tacc[1] * inv, t2 = tacc[2] * inv;
    augA[(size_t)b * N + n] = make_float4(s0, s1, s2, 1.0f);
    float b2 = t0 * t0 + t1 * t1 + t2 * t2;
    augB[(size_t)b * N + n] = make_float4(-2.f * t0, -2.f * t1, -2.f * t2, b2);
}

// ---------------------------------------------------------------------------
// Nearest-neighbor via V_WMMA_F32_16X16X4_F32, 2-deep software pipeline.
// One wave32 per (batch, 16-row tile of A). Iterates 16-col tiles of B two at
// a time: loads for tiles t+2/t+3 are issued before the WMMAs consuming
// t/t+1, giving a full iteration (2 WMMAs + 2 argmin updates) of distance
// between load issue and s_wait, with 2 loads in flight (in-order return so
// the compiler can wait at loadcnt<=2). A WGP-scope global_prefetch_b8 runs
// ~4KB ahead to pre-fill the WGP cache from L2.
// A layout (16x4 f32): lanes 0-15 -> M=lane, VGPR0=K0,VGPR1=K1;
//                      lanes 16-31 -> M=lane-16, VGPR0=K2,VGPR1=K3.
// C/D layout (16x16 f32, 8 VGPRs): VGPR r = row r (lanes 0-15, N=lane)
//                                  and row r+8 (lanes 16-31, N=lane-16).
// ---------------------------------------------------------------------------
__global__ void nn_wmma_kernel(const float4* __restrict__ augA,
                               const float4* __restrict__ augB,
                               int* __restrict__ nearest, int N)
{
    int lane = threadIdx.x & 31;
    int gw = (blockIdx.x * blockDim.x + threadIdx.x) >> 5;
    int tiles = N >> 4;              // 256 or 32: always even
    int b = gw / tiles;
    int rt = gw - b * tiles;
    int rowbase = rt << 4;
    int half = lane >> 4;            // 0: K0/K1, 1: K2/K3
    int lm = lane & 15;              // row (A) / col (B) within tile

    const float* ap = (const float*)&augA[(size_t)b * N + rowbase + lm];
    v2f a;
    a[0] = ap[half * 2 + 0];
    a[1] = ap[half * 2 + 1];

    float minv[8];
    int mini[8];
#pragma unroll
    for (int r = 0; r < 8; ++r) { minv[r] = FLT_MAX; mini[r] = 0; }

    const float4* bbase = &augB[(size_t)b * N];
    auto loadB = [&](int t) -> v2f {
        const float* bp = (const float*)&bbase[(t << 4) + lm];
        v2f bv;
        bv[0] = bp[half * 2 + 0];
        bv[1] = bp[half * 2 + 1];
        return bv;
    };
    auto update = [&](const v8f& dacc, int col) {
#pragma unroll
        for (int r = 0; r < 8; ++r) {
            if (dacc[r] < minv[r]) { minv[r] = dacc[r]; mini[r] = col; }
        }
    };

    v2f b0 = loadB(0);
    v2f b1 = loadB(1);
    for (int t = 0; t < tiles; t += 2) {
        int t2 = (t + 2 < tiles) ? t + 2 : 0;
        int t3 = (t + 3 < tiles) ? t + 3 : 1;
        v2f b2 = loadB(t2);          // in flight across the two WMMAs below
        v2f b3 = loadB(t3);
        int tp = (t + 16 < tiles) ? t + 16 : tiles - 1;
        __builtin_prefetch((const void*)((const float*)&bbase[(tp << 4) + lm]),
                           0, 3);    // WGP-scope global_prefetch_b8

        v8f c0 = {};
        v8f d0 = __builtin_amdgcn_wmma_f32_16x16x4_f32(
            false, a, false, b0, (short)0, c0, false, false);
        update(d0, (t << 4) + lm);

        v8f c1 = {};
        v8f d1 = __builtin_amdgcn_wmma_f32_16x16x4_f32(
            false, a, false, b1, (short)0, c1, false, false);
        update(d1, ((t + 1) << 4) + lm);

        b0 = b2;
        b1 = b3;
    }

    // Butterfly reduction across the 16 lanes of each half-wave (masks 1..8
    // never cross the lane-16 boundary). Tie-break on lowest column index to
    // match jnp.argmin first-occurrence semantics.
#pragma unroll
    for (int off = 1; off < 16; off <<= 1) {
#pragma unroll
        for (int r = 0; r < 8; ++r) {
            float ov = __shfl_xor(minv[r], off, 32);
            int oi = __shfl_xor(mini[r], off, 32);
            if (ov < minv[r] || (ov == minv[r] && oi < mini[r])) {
                minv[r] = ov;
                mini[r] = oi;
            }
        }
    }

    if (lm == 0) {
#pragma unroll
        for (int r = 0; r < 8; ++r)
            nearest[(size_t)b * N + rowbase + half * 8 + r] = mini[r];
    }
}

// ---------------------------------------------------------------------------
// Cosine similarity over 32-dim descriptors at gathered NN index;
// accumulate sum(cos) per stage with an LDS tree + one atomicAdd per block.
// ---------------------------------------------------------------------------
__global__ void cos_loss_kernel(const float* __restrict__ sd,
                                const float* __restrict__ td,
                                const int* __restrict__ nearest,
                                int N, float* __restrict__ accum)
{
    __shared__ float red[256];
    int idx = blockIdx.x * blockDim.x + threadIdx.x;
    int b = idx / N;
    int n = idx - b * N;
    int nn = nearest[(size_t)b * N + n];

    const float* ps = sd + (size_t)b * 32 * N + n;
    const float* pt = td + (size_t)b * 32 * N + nn;
    float num = 0.f, na = 0.f, nb = 0.f;
#pragma unroll
    for (int c = 0; c < 32; ++c) {
        float x = ps[(size_t)c * N];
        float y = pt[(size_t)c * N];
        num += x * y;
        na += x * x;
        nb += y * y;
    }
    float cosv = num / (fmaxf(sqrtf(na), CDNA5_EPS) * fmaxf(sqrtf(nb), CDNA5_EPS));

    red[threadIdx.x] = cosv;
    __syncthreads();
    for (int s = 128; s > 0; s >>= 1) {
        if (threadIdx.x < (unsigned)s) red[threadIdx.x] += red[threadIdx.x + s];
        __syncthreads();
    }
    if (threadIdx.x == 0) atomicAdd(accum, red[0]);
}

__global__ void finalize_kernel(const float* __restrict__ accum,
                                float* __restrict__ out)
{
    // loss_s = 1 - sum_cos_s / (B*N_s); out = (l0 + l1)/2
    out[0] = 1.0f - 0.5f * (accum[0] * (1.0f / 16384.0f) +
                            accum[1] * (1.0f / 2048.0f));
}

extern "C" void kernel_launch(void* const* d_in, const int* in_sizes, int n_in,
                              void* d_out, int out_size, void* d_ws, size_t ws_size,
                              hipStream_t stream) {
    (void)in_sizes; (void)n_in; (void)out_size; (void)ws_size;

    const float* csrc = (const float*)d_in[0]; // [4,3,64,64,64]
    const float* ctgt = (const float*)d_in[1];
    const float* sd0  = (const float*)d_in[2]; // [4,32,16,16,16]
    const float* td0  = (const float*)d_in[3];
    const float* sd1  = (const float*)d_in[4]; // [4,32,8,8,8]
    const float* td1  = (const float*)d_in[5];

    const int B = 4, N0 = 4096, N1 = 512;

    char* ws = (char*)d_ws;
    float4* augA0 = (float4*)(ws + 0);        // 4*4096*16 = 262144 B
    float4* augB0 = (float4*)(ws + 262144);   // 262144 B
    float4* augA1 = (float4*)(ws + 524288);   // 4*512*16 = 32768 B
    float4* augB1 = (float4*)(ws + 557056);   // 32768 B
    int*    nn0   = (int*)  (ws + 589824);    // 16384*4 = 65536 B
    int*    nn1   = (int*)  (ws + 655360);    // 2048*4  = 8192 B
    float*  acc   = (float*)(ws + 663552);    // 2 floats

    hipMemsetAsync(acc, 0, 2 * sizeof(float), stream);

    // Antialiased stage resizes (scale 4 -> 16^3, scale 8 -> 8^3)
    resize_aug_kernel<<<(B * N0) / 256, 256, 0, stream>>>(csrc, ctgt, augA0, augB0, 16, 4);
    resize_aug_kernel<<<(B * N1) / 256, 256, 0, stream>>>(csrc, ctgt, augA1, augB1, 8, 8);

    // NN search: one wave per 16-row tile; grids are exact multiples so every
    // wave is fully active (WMMA requires EXEC all-ones).
    nn_wmma_kernel<<<(B * (N0 / 16) * 32) / 128, 128, 0, stream>>>(augA0, augB0, nn0, N0);
    nn_wmma_kernel<<<(B * (N1 / 16) * 32) / 128, 128, 0, stream>>>(augA1, augB1, nn1, N1);

    cos_loss_kernel<<<(B * N0) / 256, 256, 0, stream>>>(sd0, td0, nn0, N0, acc + 0);
    cos_loss_kernel<<<(B * N1) / 256, 256, 0, stream>>>(sd1, td1, nn1, N1, acc + 1);

    finalize_kernel<<<1, 1, 0, stream>>>(acc, (float*)d_out);
}